// SSMCore_55903294324784
// MI455X (gfx1250) — compile-verified
//
#include <hip/hip_runtime.h>
#include <hip/hip_bf16.h>

#define D_MODEL 1024
#define D_STATE 16
#define D_CONV  3
#define D_INNER 2048
#define SEQ     2048
#define KCONV   (D_CONV * D_MODEL)   // 3072

typedef __attribute__((ext_vector_type(2))) float v2f;
typedef __attribute__((ext_vector_type(8))) float v8f;

__device__ __forceinline__ float sigmoidf_(float x) {
    return 1.0f / (1.0f + __expf(-x));
}

enum { EPI_BIAS = 0, EPI_SIG = 1, EPI_GATE = 2 };

// ---------------------------------------------------------------------------
// fp32 WMMA GEMM: C[M x N] = A[M x K] * Bp[K x N] (+bias, epilogue).
// One wave computes a (16*MT) x (16*NT) tile with V_WMMA_F32_16X16X4_F32,
// reusing each B fragment across MT A fragments (MT*NT independent
// accumulators -> no WMMA D->C RAW chains inside a k-chunk).
//
// B is PRE-PACKED pair-interleaved:  Bp[(k>>1)*2N + 2n + (k&1)] = B[k][n],
// so each lane's B fragment {B[k0+2hi][n], B[k0+2hi+1][n]} is one b64 load.
// The K loop is software-pipelined one stage deep: fragments for k0+4 are
// loaded before the k0 WMMAs issue, overlapping s_wait_loadcnt with math.
//
// f32 fragment layout (ISA 7.12.2):
//   A 16x4 : lane l16 = M row; vgpr0/1 = K (2hi / 2hi+1)
//   B 4x16 : lane l16 = N col; vgpr0/1 = K (2hi / 2hi+1)
//   C 16x16: vgpr i -> M = i + 8*hi, N = l16
// ---------------------------------------------------------------------------
template <int MT, int NT, int EPI>
__global__ __launch_bounds__(32)
void gemm_wmma_f32(const float* __restrict__ A, int lda,
                   const float* __restrict__ Bp, int N,
                   float* __restrict__ C, int ldc,
                   int K,
                   const float* __restrict__ bias,
                   const float* __restrict__ aux)
{
    const int lane = threadIdx.x & 31;
    const int l16  = lane & 15;
    const int hi   = lane >> 4;
    const int tileM = blockIdx.y * (16 * MT);
    const int tileN = blockIdx.x * (16 * NT);

    v8f acc[MT][NT];
#pragma unroll
    for (int mi = 0; mi < MT; ++mi)
#pragma unroll
        for (int j = 0; j < NT; ++j)
#pragma unroll
            for (int i = 0; i < 8; ++i) acc[mi][j][i] = 0.0f;

    const float* Arow = A + (size_t)(tileM + l16) * lda + 2 * hi;
    // lane's packed-B base: pair row = hi, column pair offset = 2*(tileN+l16)
    const float* Brow = Bp + (size_t)hi * 2 * N + 2 * (tileN + l16);

    // prologue: fragments for k0 = 0
    v2f a[MT], b[NT];
#pragma unroll
    for (int mi = 0; mi < MT; ++mi)
        a[mi] = *reinterpret_cast<const v2f*>(Arow + (size_t)mi * 16 * lda);
#pragma unroll
    for (int j = 0; j < NT; ++j)
        b[j] = *reinterpret_cast<const v2f*>(Brow + j * 32);

    for (int k0 = 4; k0 < K; k0 += 4) {
        // prefetch k0 fragments (consumed next trip)
        v2f a_n[MT], b_n[NT];
#pragma unroll
        for (int mi = 0; mi < MT; ++mi)
            a_n[mi] = *reinterpret_cast<const v2f*>(Arow + (size_t)mi * 16 * lda + k0);
        const float* bp = Brow + (size_t)k0 * N;
#pragma unroll
        for (int j = 0; j < NT; ++j)
            b_n[j] = *reinterpret_cast<const v2f*>(bp + j * 32);
        // math on previously-loaded fragments
#pragma unroll
        for (int mi = 0; mi < MT; ++mi)
#pragma unroll
            for (int j = 0; j < NT; ++j)
                acc[mi][j] = __builtin_amdgcn_wmma_f32_16x16x4_f32(
                    false, a[mi], false, b[j], (short)0, acc[mi][j], false, false);
#pragma unroll
        for (int mi = 0; mi < MT; ++mi) a[mi] = a_n[mi];
#pragma unroll
        for (int j = 0; j < NT; ++j) b[j] = b_n[j];
    }
#pragma unroll
    for (int mi = 0; mi < MT; ++mi)
#pragma unroll
        for (int j = 0; j < NT; ++j)
            acc[mi][j] = __builtin_amdgcn_wmma_f32_16x16x4_f32(
                false, a[mi], false, b[j], (short)0, acc[mi][j], false, false);

#pragma unroll
    for (int mi = 0; mi < MT; ++mi)
#pragma unroll
        for (int j = 0; j < NT; ++j) {
            const int n = tileN + j * 16 + l16;
            const float bv = bias ? bias[n] : 0.0f;
#pragma unroll
            for (int i = 0; i < 8; ++i) {
                const int m = tileM + mi * 16 + i + 8 * hi;
                const size_t idx = (size_t)m * ldc + n;
                float v = acc[mi][j][i] + bv;
                if (EPI == EPI_SIG)       v = sigmoidf_(v);
                else if (EPI == EPI_GATE) v = aux[idx] * sigmoidf_(v);
                C[idx] = v;
            }
        }
}

// ---------------------------------------------------------------------------
// Prep kernels
// ---------------------------------------------------------------------------
__global__ void pad_x_kernel(const float* __restrict__ x, float* __restrict__ xpad)
{
    int idx = blockIdx.x * blockDim.x + threadIdx.x;
    const int total = (SEQ + 2) * D_MODEL;
    if (idx < total) {
        int row = idx / D_MODEL;
        xpad[idx] = (row < 2) ? 0.0f : x[idx - 2 * D_MODEL];
    }
}

// conv_w (O,I,3) -> pair-interleaved B layout over r = kc*1024 + dm, n = o:
// Bp[(r>>1)*2N + 2n + (r&1)] = conv_w[n*KCONV + dm*3 + kc]
__global__ void transpose_convw_kernel(const float* __restrict__ cw, float* __restrict__ Bp)
{
    int idx = blockIdx.x * blockDim.x + threadIdx.x;
    const int total = KCONV * D_INNER;
    if (idx < total) {
        int n  = idx & (D_INNER - 1);
        int r  = idx >> 11;                 // / 2048
        int dm = r & (D_MODEL - 1);
        int kc = r >> 10;                   // / 1024
        Bp[(size_t)(r >> 1) * (2 * D_INNER) + 2 * n + (r & 1)] =
            cw[(size_t)n * KCONV + dm * D_CONV + kc];
    }
}

// Generic row-major B[k][n] (row stride ldb) -> pair-interleaved pack.
__global__ void pack_b_pairs_kernel(const float* __restrict__ B, int ldb,
                                    float* __restrict__ Bp, int K, int N)
{
    int idx = blockIdx.x * blockDim.x + threadIdx.x;
    if (idx < K * N) {
        int k = idx / N;
        int n = idx - k * N;
        Bp[(size_t)(k >> 1) * (2 * N) + 2 * n + (k & 1)] = B[(size_t)k * ldb + n];
    }
}

__global__ void softplus_A_kernel(const float* __restrict__ A_log, float* __restrict__ Av)
{
    int idx = blockIdx.x * blockDim.x + threadIdx.x;
    if (idx < D_INNER * D_STATE) {
        float v = A_log[idx];
        Av[idx] = (v > 20.0f) ? v : log1pf(__expf(v));
    }
}

__global__ __launch_bounds__(256)
void dt_kernel(const float* __restrict__ xc, float* __restrict__ dtv)
{
    __shared__ float red[256];
    const int s = blockIdx.x;
    float acc = 0.0f;
    for (int j = threadIdx.x; j < D_INNER; j += 256)
        acc += xc[(size_t)s * D_INNER + j];
    red[threadIdx.x] = acc;
    __syncthreads();
    for (int st = 128; st > 0; st >>= 1) {
        if (threadIdx.x < st) red[threadIdx.x] += red[threadIdx.x + st];
        __syncthreads();
    }
    if (threadIdx.x == 0)
        dtv[s] = sigmoidf_(red[0] * (1.0f / D_INNER)) + 0.001f;
}

// ---------------------------------------------------------------------------
// Selective scan: one thread per channel d, h[16] in registers.
// Reads u (= x_conv*gate) and overwrites it in place with ys * sigmoid(z).
// ---------------------------------------------------------------------------
__global__ __launch_bounds__(256)
void scan_kernel(float* u_y,                           // in: u, out: ys*zsig (in place)
                 const float* __restrict__ zsig,
                 const float* __restrict__ bcv,        // [SEQ][32] : B(0..15), C(16..31)
                 const float* __restrict__ dtv,        // [SEQ]
                 const float* __restrict__ Av,         // [D_INNER][16] softplus(A_log)
                 const float* __restrict__ Dv)         // [D_INNER]
{
    const int d = blockIdx.x * 256 + threadIdx.x;
    float h[D_STATE], Ad[D_STATE];
#pragma unroll
    for (int st = 0; st < D_STATE; ++st) {
        h[st] = 0.0f;
        Ad[st] = Av[d * D_STATE + st];
    }
    const float Dd = Dv[d];

    for (int s = 0; s < SEQ; ++s) {
        const float dts = dtv[s];                       // uniform -> scalar
        const float us  = u_y[(size_t)s * D_INNER + d];
        const float* bcs = bcv + s * 32;                // uniform -> scalar
        float y = Dd * us;
#pragma unroll
        for (int st = 0; st < D_STATE; ++st) {
            float ab = __expf(Ad[st] * dts);
            float hb = ab * h[st] + (ab * bcs[st] * dts) * us;
            h[st] = hb;
            y += hb * bcs[D_STATE + st];
        }
        u_y[(size_t)s * D_INNER + d] = y * zsig[(size_t)s * D_INNER + d];
    }
}

// ---------------------------------------------------------------------------
// Host-side orchestration
// ---------------------------------------------------------------------------
extern "C" void kernel_launch(void* const* d_in, const int* in_sizes, int n_in,
                              void* d_out, int out_size, void* d_ws, size_t ws_size,
                              hipStream_t stream)
{
    const float* x       = (const float*)d_in[0];
    const float* in_w    = (const float*)d_in[1];
    const float* in_b    = (const float*)d_in[2];
    const float* conv_w  = (const float*)d_in[3];
    const float* conv_b  = (const float*)d_in[4];
    const float* A_log   = (const float*)d_in[5];
    const float* Dvec    = (const float*)d_in[6];
    const float* xproj_w = (const float*)d_in[7];
    const float* xproj_b = (const float*)d_in[8];
    const float* gate_w  = (const float*)d_in[9];
    const float* gate_b  = (const float*)d_in[10];
    const float* oproj_w = (const float*)d_in[11];
    const float* oproj_b = (const float*)d_in[12];
    float* outp = (float*)d_out;

    // Workspace layout (floats), regions reused across pipeline phases:
    //   [0        .. 2099200 )  xpad            -> bcv/dtv/Av      (after conv)
    //   [2099200  .. 8390656 )  Bc (conv pack)  -> zsig + inwp     (after conv)
    //   [8390656  .. 12584960)  x_conv
    //   [12584960 .. 16779264)  u  -> ys*sigmoid(z)  (in place)
    //   [16779264 .. 20973568)  gate_w packed
    //   [20973568 .. 23070720)  o_proj_w packed
    //   [23070720 .. 23136256)  x_proj_w packed
    float* ws     = (float*)d_ws;
    float* xpad   = ws;
    float* Bc     = ws + 2099200;
    float* xconv  = ws + 8390656;
    float* ubuf   = ws + 12584960;
    float* gatep  = ws + 16779264;
    float* oprojp = ws + 20973568;
    float* xprojp = ws + 23070720;
    float* bcv    = xpad;                 // 65536   (xpad dead after conv GEMM)
    float* dtv    = xpad + 65536;         // 2048
    float* Av     = xpad + 67584;         // 32768
    float* zsig   = Bc;                   // 4194304 (Bc dead after conv GEMM)
    float* inwp   = Bc + 4194304;         // 2097152 (tail of old Bc region)

    // 1. causal pad of x
    {
        int total = (SEQ + 2) * D_MODEL;
        pad_x_kernel<<<(total + 255) / 256, 256, 0, stream>>>(x, xpad);
    }
    // 2. conv_w -> packed B [3072 x 2048]
    {
        int total = KCONV * D_INNER;
        transpose_convw_kernel<<<(total + 255) / 256, 256, 0, stream>>>(conv_w, Bc);
    }
    // 3. pack gate_w / o_proj_w / x_proj_w
    pack_b_pairs_kernel<<<(D_INNER * D_INNER + 255) / 256, 256, 0, stream>>>(
        gate_w, D_INNER, gatep, D_INNER, D_INNER);
    pack_b_pairs_kernel<<<(D_INNER * D_MODEL + 255) / 256, 256, 0, stream>>>(
        oproj_w, D_MODEL, oprojp, D_INNER, D_MODEL);
    pack_b_pairs_kernel<<<(D_INNER * 2 * D_STATE + 255) / 256, 256, 0, stream>>>(
        xproj_w, 2 * D_STATE, xprojp, D_INNER, 2 * D_STATE);

    // 4. conv as GEMM: x_conv = xpad(view, K=3072, lda=1024) * Bc + conv_b
    gemm_wmma_f32<2, 4, EPI_BIAS><<<dim3(D_INNER / 64, SEQ / 32), 32, 0, stream>>>(
        xpad, D_MODEL, Bc, D_INNER, xconv, D_INNER, KCONV, conv_b, nullptr);

    // 5. A = softplus(A_log); pack z-half of in_proj_w (both into dead regions)
    softplus_A_kernel<<<(D_INNER * D_STATE + 255) / 256, 256, 0, stream>>>(A_log, Av);
    pack_b_pairs_kernel<<<(D_MODEL * D_INNER + 255) / 256, 256, 0, stream>>>(
        in_w + D_INNER, 2 * D_INNER, inwp, D_MODEL, D_INNER);

    // 6. zsig = sigmoid(x @ in_proj_w[:, D_INNER:] + b)
    gemm_wmma_f32<2, 4, EPI_SIG><<<dim3(D_INNER / 64, SEQ / 32), 32, 0, stream>>>(
        x, D_MODEL, inwp, D_INNER, zsig, D_INNER, D_MODEL, in_b + D_INNER, nullptr);

    // 7. u = x_conv * sigmoid(x_conv @ gate_w + gate_b)
    gemm_wmma_f32<2, 4, EPI_GATE><<<dim3(D_INNER / 64, SEQ / 32), 32, 0, stream>>>(
        xconv, D_INNER, gatep, D_INNER, ubuf, D_INNER, D_INNER, gate_b, xconv);

    // 8. bc = x_conv @ x_proj_w + x_proj_b   (N = 32)
    gemm_wmma_f32<2, 2, EPI_BIAS><<<dim3(1, SEQ / 32), 32, 0, stream>>>(
        xconv, D_INNER, xprojp, 2 * D_STATE, bcv, 2 * D_STATE, D_INNER,
        xproj_b, nullptr);

    // 9. dt[s] = sigmoid(mean(x_conv[s,:])) + 0.001
    dt_kernel<<<SEQ, 256, 0, stream>>>(xconv, dtv);

    // 10. selective scan; ubuf becomes ys * sigmoid(z)
    scan_kernel<<<D_INNER / 256, 256, 0, stream>>>(ubuf, zsig, bcv, dtv, Av, Dvec);

    // 11. out = (ys * zsig) @ o_proj_w + o_proj_b
    gemm_wmma_f32<2, 4, EPI_BIAS><<<dim3(D_MODEL / 64, SEQ / 32), 32, 0, stream>>>(
        ubuf, D_INNER, oprojp, D_MODEL, outp, D_MODEL, D_INNER, oproj_b, nullptr);
}